// GNN_34883724378711
// MI455X (gfx1250) — compile-verified
//
#include <hip/hip_runtime.h>
#include <hip/hip_bf16.h>
#include <cstdint>

typedef __attribute__((ext_vector_type(2))) float v2f;
typedef __attribute__((ext_vector_type(8))) float v8f;

#define D_IN  128
#define D_HID 48
#define D_OUT 32

// ---------------- utility kernels ----------------

__global__ void gcn_zero(float* p, long long n) {
    long long i = (long long)blockIdx.x * blockDim.x + threadIdx.x;
    if (i < n) p[i] = 0.0f;
}

__global__ void gcn_degree(const long long* __restrict__ dst,
                           const float* __restrict__ ew,
                           float* __restrict__ deg, int E) {
    int e = blockIdx.x * blockDim.x + threadIdx.x;
    if (e < E) atomicAdd(&deg[dst[e]], ew[e]);
}

__global__ void gcn_dinv(float* __restrict__ deg, int N) {
    int i = blockIdx.x * blockDim.x + threadIdx.x;
    if (i < N) deg[i] = rsqrtf(deg[i] + 1.0f);   // deg+1 >= 1 always
}

// ---------------- WMMA fp32 GEMM: Y[nrows x NDIM] = X[nrows x KDIM] @ W[KDIM x NDIM] ----------------
// One wave computes a 16 x NDIM tile via V_WMMA_F32_16X16X4_F32, 8 waves per block.
// W is staged transposed in LDS so B fragments are contiguous ds_load_b64.

template <int KDIM, int NDIM>
__global__ __launch_bounds__(256) void gcn_gemm_wmma(const float* __restrict__ X,
                                                     const float* __restrict__ W,
                                                     float* __restrict__ Y,
                                                     int nrows) {
    constexpr int NT = NDIM / 16;
    __shared__ float wT[NDIM * KDIM];   // wT[n*KDIM + k] = W[k*NDIM + n]

    int tid = threadIdx.x;
    for (int t = tid; t < KDIM * NDIM; t += 256) {
        int k = t / NDIM, n = t % NDIM;
        wT[n * KDIM + k] = W[t];
    }
    __syncthreads();

    int wave = tid >> 5;
    int lane = tid & 31;
    int lh   = lane >> 4;      // 0 for lanes 0-15, 1 for 16-31
    int l15  = lane & 15;
    int kh   = lh << 1;        // K sub-offset: 0 or 2 (per 16x4 A / 4x16 B VGPR layout)

    int rowbase = blockIdx.x * 128 + wave * 16;
    int rowL = rowbase + l15;
    if (rowL > nrows - 1) rowL = nrows - 1;   // clamp loads; stores are guarded
    const float* xrow = X + (size_t)rowL * KDIM;

    v8f c[NT] = {};   // fp32 accumulators, 8 VGPRs each

    #pragma unroll
    for (int k = 0; k < KDIM; k += 4) {
        // A 16x4 tile: lane<16 holds (M=l15, K=k..k+1); lane>=16 holds K=k+2..k+3
        v2f a = *(const v2f*)(xrow + k + kh);
        #pragma unroll
        for (int nt = 0; nt < NT; ++nt) {
            // B 4x16 tile: b.x = W[k+kh][ncol], b.y = W[k+kh+1][ncol]
            v2f b = *(const v2f*)(&wT[(nt * 16 + l15) * KDIM + k + kh]);
            c[nt] = __builtin_amdgcn_wmma_f32_16x16x4_f32(
                        false, a, false, b, (short)0, c[nt], false, false);
        }
    }

    // C/D layout: VGPR i -> M = i + (lane>=16 ? 8 : 0), N = lane&15
    #pragma unroll
    for (int nt = 0; nt < NT; ++nt) {
        int ncol = nt * 16 + l15;
        #pragma unroll
        for (int i = 0; i < 8; ++i) {
            int r = rowbase + i + (lh << 3);
            if (r < nrows) Y[(size_t)r * NDIM + ncol] = c[nt][i];
        }
    }
}

// ---------------- edge scatter: agg[dst] += dinv[src]*ew*dinv[dst] * h[src] ----------------

// layer 1: 48 features, 16 lanes per edge, 3 features each (f, f+16, f+32)
__global__ void gcn_scatter48(const long long* __restrict__ src,
                              const long long* __restrict__ dst,
                              const float* __restrict__ ew,
                              const float* __restrict__ dinv,
                              const float* __restrict__ h,
                              float* __restrict__ agg, int E) {
    int t = blockIdx.x * 256 + threadIdx.x;
    int e = t >> 4;
    int f = t & 15;
    if (e >= E) return;
    long long s = src[e], d = dst[e];
    float norm = dinv[s] * ew[e] * dinv[d];
    const float* hs = h + (size_t)s * D_HID;
    float* ad = agg + (size_t)d * D_HID;
    #pragma unroll
    for (int j = 0; j < 3; ++j) {
        int col = f + j * 16;
        atomicAdd(&ad[col], norm * hs[col]);
    }
}

// layer 2: 32 features, one wave per edge, one feature per lane
__global__ void gcn_scatter32(const long long* __restrict__ src,
                              const long long* __restrict__ dst,
                              const float* __restrict__ ew,
                              const float* __restrict__ dinv,
                              const float* __restrict__ h,
                              float* __restrict__ agg, int E) {
    int t = blockIdx.x * 256 + threadIdx.x;
    int e = t >> 5;
    int f = t & 31;
    if (e >= E) return;
    long long s = src[e], d = dst[e];
    float norm = dinv[s] * ew[e] * dinv[d];
    atomicAdd(&agg[(size_t)d * D_OUT + f], norm * h[(size_t)s * D_OUT + f]);
}

// ---------------- finalize: agg + dinv^2 * h + bias (+ReLU for layer 1) ----------------

__global__ void gcn_finalize_relu48(float* __restrict__ agg,
                                    const float* __restrict__ h,
                                    const float* __restrict__ dinv,
                                    const float* __restrict__ bias, int total) {
    int t = blockIdx.x * blockDim.x + threadIdx.x;
    if (t >= total) return;
    int i = t / D_HID, f = t - i * D_HID;
    float di = dinv[i];
    float v = agg[t] + di * di * h[t] + bias[f];
    agg[t] = v > 0.0f ? v : 0.0f;   // in place; becomes GEMM2 input
}

__global__ void gcn_finalize32(const float* __restrict__ agg,
                               const float* __restrict__ h,
                               const float* __restrict__ dinv,
                               const float* __restrict__ bias,
                               float* __restrict__ out, int total) {
    int t = blockIdx.x * blockDim.x + threadIdx.x;
    if (t >= total) return;
    int i = t / D_OUT, f = t - i * D_OUT;
    float di = dinv[i];
    out[t] = agg[t] + di * di * h[t] + bias[f];
}

// ---------------- launcher ----------------

extern "C" void kernel_launch(void* const* d_in, const int* in_sizes, int n_in,
                              void* d_out, int out_size, void* d_ws, size_t ws_size,
                              hipStream_t stream) {
    (void)n_in; (void)out_size; (void)ws_size;

    const float*     x   = (const float*)d_in[0];
    const long long* ei  = (const long long*)d_in[1];   // int64 [2, E]
    const float*     ew  = (const float*)d_in[2];
    const float*     W1  = (const float*)d_in[3];       // [128, 48]
    const float*     b1  = (const float*)d_in[4];
    const float*     W2  = (const float*)d_in[5];       // [48, 32]
    const float*     b2  = (const float*)d_in[6];
    float*           out = (float*)d_out;               // [N, 32]

    const int N = in_sizes[0] / D_IN;
    const int E = in_sizes[2];
    const long long* src = ei;
    const long long* dst = ei + E;

    // workspace layout (floats): [deg N][agg1 N*48][agg2 N*32][h1/h2 N*48]
    float* deg  = (float*)d_ws;
    float* agg1 = deg  + N;
    float* agg2 = agg1 + (size_t)N * D_HID;
    float* h1   = agg2 + (size_t)N * D_OUT;
    float* h2   = h1;   // reused after finalize1 consumed h1

    // zero deg + agg1 + agg2 (contiguous) every call -> deterministic replays
    long long zcount = (long long)N * (1 + D_HID + D_OUT);
    gcn_zero<<<(int)((zcount + 255) / 256), 256, 0, stream>>>(deg, zcount);

    gcn_degree<<<(E + 255) / 256, 256, 0, stream>>>(dst, ew, deg, E);
    gcn_dinv<<<(N + 255) / 256, 256, 0, stream>>>(deg, N);   // deg -> dinv in place

    // layer 1
    gcn_gemm_wmma<D_IN, D_HID><<<(N + 127) / 128, 256, 0, stream>>>(x, W1, h1, N);
    gcn_scatter48<<<(E + 15) / 16, 256, 0, stream>>>(src, dst, ew, deg, h1, agg1, E);
    gcn_finalize_relu48<<<(N * D_HID + 255) / 256, 256, 0, stream>>>(agg1, h1, deg, b1, N * D_HID);

    // layer 2
    gcn_gemm_wmma<D_HID, D_OUT><<<(N + 127) / 128, 256, 0, stream>>>(agg1, W2, h2, N);
    gcn_scatter32<<<(E + 7) / 8, 256, 0, stream>>>(src, dst, ew, deg, h2, agg2, E);
    gcn_finalize32<<<(N * D_OUT + 255) / 256, 256, 0, stream>>>(agg2, h2, deg, b2, out, N * D_OUT);
}